// spatial_attention_27144193311035
// MI455X (gfx1250) — compile-verified
//
#include <hip/hip_runtime.h>

typedef __attribute__((ext_vector_type(2))) float v2f;
typedef __attribute__((ext_vector_type(4))) float v4f;
typedef __attribute__((ext_vector_type(8))) float v8f;

#define NB 128   // batches
#define NN 512   // pedestrians
#define ND 64    // hidden dim
#define WSTRIDE 516  // 512 + 4 pad -> 4-bank rotation per row, conflict-free for WMMA A reads

__global__ __launch_bounds__(256)
void spatial_attn_fused(const float* __restrict__ hidden,
                        const float* __restrict__ dist,
                        const float* __restrict__ bear,
                        const float* __restrict__ head,
                        const float* __restrict__ mask,
                        const float* __restrict__ domain,
                        float* __restrict__ out)
{
    __shared__ float sW[16 * WSTRIDE];   // 16 weight rows (padded)
    __shared__ float sRed[4 * 256];      // partial-C exchange (4 col tiles)
    __shared__ float sDom[144];
    __shared__ float sMask[NN];

    const int tid  = threadIdx.x;
    const int lane = tid & 31;
    const int wave = tid >> 5;
    const int b    = blockIdx.x >> 5;        // 32 row-tiles per batch
    const int i0   = (blockIdx.x & 31) << 4; // first row of this tile

    const float* Hb = hidden + (size_t)b * (NN * ND);

    // ---- stage domain + sequence-mask row; emit hidden passthrough half ----
    if (tid < 144) sDom[tid] = domain[tid];
    for (int t = tid; t < NN; t += 256) sMask[t] = mask[(size_t)b * NN + t];
    {   // 16 rows x 64 cols = 256 float4, one per thread
        const int r  = tid >> 4;
        const int c4 = tid & 15;
        v4f h4 = *(const v4f*)&Hb[(size_t)(i0 + r) * ND + c4 * 4];
        *(v4f*)&out[((size_t)(b * NN + i0 + r)) * (2 * ND) + ND + c4 * 4] = h4;
    }
    __syncthreads();

    // ---- Phase A: masked-softmax weights for rows i0..i0+15 ----
    // lane owns j = (lane + m*32)*4 + c  (m=0..3, c=0..3) -> b128 coalesced loads
    #pragma unroll
    for (int rr = 0; rr < 2; ++rr) {
        const int i = i0 + wave * 2 + rr;
        const size_t rowOff = ((size_t)b * NN + i) * NN;
        const v4f* dR = (const v4f*)(dist + rowOff);
        const v4f* bR = (const v4f*)(bear + rowOff);
        const v4f* hR = (const v4f*)(head + rowOff);
        const float smi = sMask[i];

        float w[16];
        #pragma unroll
        for (int m = 0; m < 4; ++m) {
            const int q = lane + (m << 5);      // float4 index within row
            v4f d4 = dR[q];
            v4f b4 = bR[q];
            v4f h4 = hR[q];
            v4f k4 = ((const v4f*)sMask)[q];
            #pragma unroll
            for (int c = 0; c < 4; ++c) {
                const int j = q * 4 + c;
                float sb = b4[c] + 15.0f; sb = (sb >= 360.0f) ? sb - 360.0f : sb;
                float sh = h4[c] + 15.0f; sh = (sh >= 360.0f) ? sh - 360.0f : sh;
                int i1 = (int)floorf(sh * (1.0f / 30.0f));
                int i2 = (int)floorf(sb * (1.0f / 30.0f));
                i1 = i1 < 0 ? 0 : (i1 > 11 ? 11 : i1);
                i2 = i2 < 0 ? 0 : (i2 > 11 ? 11 : i2);
                float wv = fmaxf(sDom[i1 * 12 + i2] - d4[c], 0.0f);
                float mm = (j == i) ? 0.0f : smi * k4[c];
                w[m * 4 + c] = wv * mm;
            }
        }
        // row max (regs then wave32 butterfly)
        float mx = 0.0f;
        #pragma unroll
        for (int k = 0; k < 16; ++k) mx = fmaxf(mx, w[k]);
        #pragma unroll
        for (int off = 16; off > 0; off >>= 1)
            mx = fmaxf(mx, __shfl_xor(mx, off, 32));
        const float inv = 1.0f / (mx + 1e-14f);
        // masked exp + sum
        float s = 0.0f;
        #pragma unroll
        for (int k = 0; k < 16; ++k) {
            float v = w[k] * inv;
            float e = (v != 0.0f) ? __expf(v) : 0.0f;
            w[k] = e;
            s += e;
        }
        #pragma unroll
        for (int off = 16; off > 0; off >>= 1)
            s += __shfl_xor(s, off, 32);
        const float dn = (s != 0.0f) ? 1.0f / (s + 1e-14f) : 0.0f;
        const int lr = wave * 2 + rr;
        #pragma unroll
        for (int m = 0; m < 4; ++m) {
            v4f o;
            o[0] = w[m * 4 + 0] * dn;
            o[1] = w[m * 4 + 1] * dn;
            o[2] = w[m * 4 + 2] * dn;
            o[3] = w[m * 4 + 3] * dn;
            *(v4f*)&sW[lr * WSTRIDE + (lane + (m << 5)) * 4] = o;  // b128, conflict-free
        }
    }
    __syncthreads();

    // ---- Phase B: (16x512)[LDS] x (512x64)[global] via v_wmma_f32_16x16x4_f32 ----
    const int ct   = wave & 3;   // 16-wide column tile
    const int kh   = wave >> 2;  // K half (256 each)
    const int half = lane >> 4;
    const int l15  = lane & 15;
    const int col  = ct * 16 + l15;

    v8f acc = {};
    const int kbeg = kh * 256;
    #pragma unroll 4
    for (int k0 = kbeg; k0 < kbeg + 256; k0 += 4) {
        const int kk = k0 + half * 2;           // ISA A layout: K0/K2 | K1/K3 lane split
        v2f a;
        a.x = sW[l15 * WSTRIDE + kk];
        a.y = sW[l15 * WSTRIDE + kk + 1];
        v2f bv;
        bv.x = Hb[(size_t)kk * ND + col];       // ISA B layout: K0/K2 | K1/K3 lane split
        bv.y = Hb[(size_t)(kk + 1) * ND + col];
        acc = __builtin_amdgcn_wmma_f32_16x16x4_f32(
                  false, a, false, bv, (short)0, acc, false, false);
    }

    // reduce the two K-halves through LDS, store weighted half of output
    if (kh == 1) {
        #pragma unroll
        for (int v = 0; v < 8; ++v) sRed[ct * 256 + v * 32 + lane] = acc[v];
    }
    __syncthreads();
    if (kh == 0) {
        #pragma unroll
        for (int v = 0; v < 8; ++v) {
            float r = acc[v] + sRed[ct * 256 + v * 32 + lane];
            int row = i0 + v + half * 8;        // C/D layout: M = v (+8 for hi lanes)
            out[((size_t)(b * NN + row)) * (2 * ND) + col] = r;
        }
    }
}

extern "C" void kernel_launch(void* const* d_in, const int* in_sizes, int n_in,
                              void* d_out, int out_size, void* d_ws, size_t ws_size,
                              hipStream_t stream) {
    const float* hidden = (const float*)d_in[0];
    const float* dist   = (const float*)d_in[1];
    const float* bear   = (const float*)d_in[2];
    const float* head   = (const float*)d_in[3];
    const float* mask   = (const float*)d_in[4];
    const float* dom    = (const float*)d_in[5];

    dim3 grid(NB * (NN / 16));   // 4096 blocks, one 16-row tile each
    spatial_attn_fused<<<grid, 256, 0, stream>>>(
        hidden, dist, bear, head, mask, dom, (float*)d_out);
}